// AbstractDTW_46712064311509
// MI455X (gfx1250) — compile-verified
//
#include <hip/hip_runtime.h>
#include <hip/hip_bf16.h>
#include <stdint.h>

#define N     1024      // sequence length (both snake and contour)
#define T     256       // threads per block (8 wave32s)
#define RPT   4         // rows per thread (T*RPT == N)
#define NDIAG (2*N - 1) // anti-diagonals

typedef __attribute__((ext_vector_type(4))) unsigned int uint4v;
typedef __attribute__((ext_vector_type(8))) int          int8v;
typedef __attribute__((ext_vector_type(4))) int          int4v;

// softmin over 3 values with gamma=1, numerically identical to
// -logsumexp(-v): shift by min, native base-2 exp/log.
__device__ __forceinline__ float softmin3(float a, float b, float c) {
    const float L2E = 1.4426950408889634f;  // log2(e)
    const float LN2 = 0.6931471805599453f;  // ln(2)
    float m = fminf(a, fminf(b, c));
    float s = __builtin_amdgcn_exp2f((m - a) * L2E)
            + __builtin_amdgcn_exp2f((m - b) * L2E)
            + __builtin_amdgcn_exp2f((m - c) * L2E);
    return m - LN2 * __builtin_amdgcn_logf(s);  // v_log_f32 is log2
}

__global__ __launch_bounds__(T) void sdtw_kernel(
    const float* __restrict__ snake,    // (B, N, 2)
    const float* __restrict__ contour,  // (B, N, 2)
    float* __restrict__ partial)        // (B,) per-batch R[N-1,N-1]
{
    __shared__ float2 c_lds[N];          // staged contour (8 KB)
    __shared__ float  bufs[3][T + 1];    // rotating top-row exchange buffers

    const int   b   = blockIdx.x;
    const int   t   = threadIdx.x;
    const float INF = __builtin_inff();

    // ---- Stage contour into LDS via Tensor Data Mover (wave 0 issues) ----
    // D#: 2D tensor, data_size=4B, tensor_dim0 = tile_dim0 = 2048 floats,
    // tensor_dim1 = tile_dim1 = 1, stride0 = 2048.  (CDNA5 ISA)
    if (t < 32) {
        uint64_t ga      = (uint64_t)(uintptr_t)(contour + (size_t)b * (2 * N));
        uint32_t lds_off = (uint32_t)(uintptr_t)&c_lds[0];
        uint4v g0;
        g0.x = 1u;                                   // count = 1 valid descriptor
        g0.y = lds_off;                              // lds_addr (bytes)
        g0.z = (uint32_t)ga;                         // global_addr[31:0]
        g0.w = ((uint32_t)(ga >> 32) & 0x01FFFFFFu)  // global_addr[56:32]
             | (2u << 30);                           // type = 2 ("image")
        int8v g1;
        g1[0] = (int)0x00020000u;   // workgroup_mask=0, data_size=2 (4 bytes)
        g1[1] = (int)(2048u << 16); // tensor_dim0 = 2048 (low 16 @ [63:48])
        g1[2] = (int)(1u << 16);    // tensor_dim0 hi=0, tensor_dim1 = 1
        g1[3] = (int)(2048u << 16); // tensor_dim1 hi=0, tile_dim0 = 2048
        g1[4] = 1;                  // tile_dim1 = 1, tile_dim2 = 0
        g1[5] = 2048;               // tensor_dim0_stride low 32
        g1[6] = (int)(2048u << 16); // stride0 hi=0, tensor_dim1_stride low16
        g1[7] = 0;
        int4v gz4 = {0, 0, 0, 0};
        int8v gz8 = {0, 0, 0, 0, 0, 0, 0, 0};
        // 6-arg form (amdgpu-toolchain / clang-23 + therock headers)
        __builtin_amdgcn_tensor_load_to_lds(g0, g1, gz4, gz4, gz8, 0);
    }
    // Redundant direct staging (identical bytes) so correctness never
    // depends on TDM descriptor semantics; benign same-value race.
    {
        const float2* cp = (const float2*)(contour + (size_t)b * (2 * N));
        #pragma unroll
        for (int k = 0; k < RPT; ++k) {
            int idx = t + k * T;
            c_lds[idx] = cp[idx];
        }
    }
    if (t < 32) __builtin_amdgcn_s_wait_tensorcnt(0);
    __syncthreads();

    // ---- Init rotating exchange buffers to +inf (after TDM completion) ----
    #pragma unroll
    for (int r = 0; r < 3; ++r) {
        bufs[r][t] = INF;
        if (t == 0) bufs[r][T] = INF;
    }

    // ---- Per-thread state: rows i = 4t..4t+3 ----
    float sx[RPT], sy[RPT];
    {
        const float2* sp = (const float2*)(snake + (size_t)b * (2 * N));
        #pragma unroll
        for (int k = 0; k < RPT; ++k) {
            float2 s = sp[RPT * t + k];
            sx[k] = s.x; sy[k] = s.y;
        }
    }

    float p1[RPT], p2[RPT];  // row value at diag d-1 / d-2
    #pragma unroll
    for (int k = 0; k < RPT; ++k) { p1[k] = INF; p2[k] = INF; }

    // Base case d=0: R[0,0] = D(0,0)   (thread 0, row 0)
    if (t == 0) {
        float2 c0 = c_lds[0];
        float dx = sx[0] - c0.x, dy = sy[0] - c0.y;
        p1[0] = dx * dx + dy * dy;
    }

    // Contour shift registers: cc[k] = contour[d - 4t - k] (valid lanes only)
    float ccx[RPT], ccy[RPT];
    #pragma unroll
    for (int k = 0; k < RPT; ++k) {
        unsigned uj = (unsigned)(1 - RPT * t - k);
        float2 c = c_lds[uj < N ? uj : 0];
        ccx[k] = c.x; ccy[k] = c.y;
    }

    __syncthreads();

    int pi2 = 0, pi1 = 1, pic = 2;   // buffers holding d-2, d-1, write target
    const int base_i = RPT * t;

    for (int d = 1; d < NDIAG; ++d) {
        float nb1 = bufs[pi1][t];    // neighbor's top row (i-1 for k==0) @ d-1
        float nb2 = bufs[pi2][t];    // @ d-2
        float cur[RPT];
        #pragma unroll
        for (int k = 0; k < RPT; ++k) {
            int   i  = base_i + k;
            float a  = (k == 0) ? nb1 : p1[k - 1];  // R[i-1, j]
            float g  = (k == 0) ? nb2 : p2[k - 1];  // R[i-1, j-1]
            float cL = p1[k];                       // R[i, j-1]
            float sm = softmin3(a, g, cL);
            float dx = sx[k] - ccx[k];
            float dy = sy[k] - ccy[k];
            float D  = dx * dx + dy * dy;
            int   j  = d - i;
            bool  active = (j >= 0) && (j < N);
            cur[k] = active ? (D + sm) : INF;       // NaN from inf-inf masked
        }
        bufs[pic][t + 1] = cur[RPT - 1];            // publish top row
        #pragma unroll
        for (int k = 0; k < RPT; ++k) { p2[k] = p1[k]; p1[k] = cur[k]; }
        // shift contour registers for next diagonal
        #pragma unroll
        for (int k = RPT - 1; k > 0; --k) { ccx[k] = ccx[k - 1]; ccy[k] = ccy[k - 1]; }
        {
            unsigned uj = (unsigned)(d + 1 - base_i);
            float2 c = c_lds[uj < N ? uj : 0];
            ccx[0] = c.x; ccy[0] = c.y;
        }
        __syncthreads();
        int tmp = pi2; pi2 = pi1; pi1 = pic; pic = tmp;
    }

    if (t == T - 1) partial[b] = p1[RPT - 1];       // R[1023, 1023]
}

__global__ void sdtw_reduce(const float* __restrict__ partial,
                            float* __restrict__ out, int B)
{
    if (threadIdx.x == 0) {
        float s = 0.0f;
        for (int i = 0; i < B; ++i) s += partial[i];   // fixed order: deterministic
        out[0] = s / (float)B;
    }
}

extern "C" void kernel_launch(void* const* d_in, const int* in_sizes, int n_in,
                              void* d_out, int out_size, void* d_ws, size_t ws_size,
                              hipStream_t stream) {
    const float* snake   = (const float*)d_in[0];
    const float* contour = (const float*)d_in[1];
    float*       out     = (float*)d_out;
    float*       partial = (float*)d_ws;

    int B = in_sizes[0] / (N * 2);   // 64 batches
    sdtw_kernel<<<B, T, 0, stream>>>(snake, contour, partial);
    sdtw_reduce<<<1, 32, 0, stream>>>(partial, out, B);
}